// _ChannelwiseEdgeConv_27754078667287
// MI455X (gfx1250) — compile-verified
//
#include <hip/hip_runtime.h>
#include <hip/hip_bf16.h>

// ---------------------------------------------------------------------------
// CDNA5 (gfx1250) wave32 WMMA f32 16x16x4 helpers
// A (16x4 f32): 2 VGPRs; lane L holds A[L%16][j + 2*(L/16)] in reg j
// B (4x16 f32): 2 VGPRs; lane L holds B[j + 2*(L/16)][L%16] in reg j
// C/D (16x16 f32): 8 VGPRs; lane L reg j holds D[j + 8*(L/16)][L%16]
// ---------------------------------------------------------------------------
typedef __attribute__((ext_vector_type(2))) float v2f;
typedef __attribute__((ext_vector_type(8))) float v8f;

__device__ __forceinline__ v8f wmma4(v2f a, v2f b, v8f c) {
  return __builtin_amdgcn_wmma_f32_16x16x4_f32(
      /*neg_a=*/false, a, /*neg_b=*/false, b,
      /*c_mod=*/(short)0, c, /*reuse_a=*/false, /*reuse_b=*/false);
}

// One 16x16 output tile; all strides compile-time so fragment addresses are
// constant-affine (lets the compiler use ds_load_2addr with immediate offsets).
template <int LDA, int KSTEP, int LDB, int K>
__device__ __forceinline__ void gemm_tile(const float* __restrict__ A,
                                          const float* __restrict__ B,
                                          v8f& acc, int lane) {
  const int M = lane & 15;
  const int half = lane >> 4;
#pragma unroll
  for (int k0 = 0; k0 < K; k0 += 4) {
    v2f a, b;
#pragma unroll
    for (int j = 0; j < 2; ++j) {
      const int k = k0 + j + 2 * half;
      a[j] = A[M * LDA + k * KSTEP];
      b[j] = B[k * LDB + M];
    }
    acc = wmma4(a, b, acc);
  }
}

// ---------------------------------------------------------------------------
// gfx1250 async global->LDS staging (ASYNCcnt path), with sync fallback.
// Builtin signature (verified by compiler diagnostics):
//   _b128: (int4 AS1*, int4 AS3*, imm offset, imm cpol)
//   _b32 : (int  AS1*, int  AS3*, imm offset, imm cpol)
// ---------------------------------------------------------------------------
#if __has_builtin(__builtin_amdgcn_global_load_async_to_lds_b128) && \
    __has_builtin(__builtin_amdgcn_global_load_async_to_lds_b32) &&  \
    __has_builtin(__builtin_amdgcn_s_wait_asynccnt)
#define ASYNC_LDS 1
#else
#define ASYNC_LDS 0
#endif

typedef int vsi4 __attribute__((vector_size(16)));

__device__ __forceinline__ void stage16(float* ldst, const float* gsrc) {
#if ASYNC_LDS
  __builtin_amdgcn_global_load_async_to_lds_b128(
      (__attribute__((address_space(1))) vsi4*)gsrc,
      (__attribute__((address_space(3))) vsi4*)ldst, 0, 0);
#else
  ldst[0] = gsrc[0]; ldst[1] = gsrc[1]; ldst[2] = gsrc[2]; ldst[3] = gsrc[3];
#endif
}

__device__ __forceinline__ void stage_wait() {
#if ASYNC_LDS
  __builtin_amdgcn_s_wait_asynccnt(0);
#endif
}

__device__ __forceinline__ float silu_act(float x) {
  // silu(x) * ACT_SCALE, ACT_SCALE = 1/sqrt(E[silu(N(0,1))^2]) ~= 1.67653
  return (x / (1.0f + expf(-x))) * 1.67653f;
}

// ---------------------------------------------------------------------------
// Kernel Z: zero the msg accumulator
// ---------------------------------------------------------------------------
__global__ void k_zero(float* __restrict__ p, int n) {
  for (int i = blockIdx.x * blockDim.x + threadIdx.x; i < n; i += gridDim.x * blockDim.x)
    p[i] = 0.0f;
}

// ---------------------------------------------------------------------------
// Kernel 1: node "up" linears.  (weights staged RAW; 1/8 applied at D store)
//   xs[n, v]      = (sum_u nf[n, u]            * Wus[u, v]) / 8
//   xv[n, v, m]   = (sum_u nf[n, 64 + u*3 + m] * Wuv[u, v]) / 8
// ---------------------------------------------------------------------------
__global__ void k_node_up(const float* __restrict__ nf,
                          const float* __restrict__ wus,
                          const float* __restrict__ wuv,
                          float* __restrict__ xs, float* __restrict__ xv, int N) {
  __shared__ float sF[16 * 256];
  __shared__ float sWs[64 * 64];
  __shared__ float sWv[64 * 64];
  const int tid = threadIdx.x;
  const int base = blockIdx.x * 16;

  // Async-stage node-feature tile (row-clamped for the tail tile) + raw weights.
  for (int v = tid; v < 1024; v += 256) {          // 16 rows * 64 float4
    const int row = v >> 6;
    const int n = (base + row < N) ? (base + row) : (N - 1);
    stage16(&sF[v * 4], &nf[(size_t)n * 256 + (v & 63) * 4]);
  }
  for (int v = tid; v < 1024; v += 256) {          // 64x64 weights
    stage16(&sWs[v * 4], &wus[v * 4]);
    stage16(&sWv[v * 4], &wuv[v * 4]);
  }
  stage_wait();
  __syncthreads();

  const int lane = tid & 31;
  const int wave = tid >> 5;
  const int Mn = lane & 15;
  const int half = lane >> 4;

  for (int t = wave * 2; t < wave * 2 + 2; ++t) {
    const int g = t >> 2;      // 0 = xs, 1..3 = xv component m = g-1 (wave-uniform)
    const int c = t & 3;       // column tile
    v8f acc = {};
    if (g == 0)
      gemm_tile<256, 1, 64, 64>(sF, &sWs[c * 16], acc, lane);
    else
      gemm_tile<256, 3, 64, 64>(&sF[64 + (g - 1)], &sWv[c * 16], acc, lane);
#pragma unroll
    for (int j = 0; j < 8; ++j) {
      const int n = base + j + 8 * half;
      if (n < N) {
        const int col = c * 16 + Mn;
        const float val = acc[j] * 0.125f;         // 1/sqrt(64)
        if (g == 0) xs[(size_t)n * 64 + col] = val;
        else        xv[(size_t)n * 192 + col * 3 + (g - 1)] = val;
      }
    }
  }
}

// ---------------------------------------------------------------------------
// Kernel 2: per-edge radial MLP (WMMA) + message build + f32 atomic scatter.
// Grid-stride over 64-edge tiles; weights async-staged RAW once per block.
// Scale folding (exact linear refactorings):
//   emb' = emb/sqrt(8)            -> h1_pre = emb' @ W0_raw
//   h1'  = silu(h1_pre)*ACT/8     -> h2_pre = h1' @ W1_raw
//   h2'  = silu(h2_pre)*ACT/8     -> gates  = h2' @ W2_raw
// ---------------------------------------------------------------------------
__global__ void k_edge(const float* __restrict__ evec, const float* __restrict__ elen,
                       const int* __restrict__ send, const int* __restrict__ recv,
                       const float* __restrict__ w0g, const float* __restrict__ w1g,
                       const float* __restrict__ w2g,
                       const float* __restrict__ xs, const float* __restrict__ xv,
                       float* __restrict__ msg, int E) {
  __shared__ float sW0[8 * 64];
  __shared__ float sW1[64 * 64];
  __shared__ float sW2[64 * 256];
  __shared__ float sEmb[64 * 8];
  __shared__ float sH1[64 * 64];
  __shared__ float sH2[64 * 64];
  __shared__ float sG[64 * 256];
  __shared__ float sY[64 * 3];
  __shared__ int   sS[64];
  __shared__ int   sR[64];

  const int tid = threadIdx.x;
  const int lane = tid & 31;
  const int wave = tid >> 5;
  const int Mn = lane & 15;
  const int half = lane >> 4;

  // Async-stage all three weight matrices raw (once per block).
  if (tid < 128) stage16(&sW0[tid * 4], &w0g[tid * 4]);      // 8x64
  for (int v = tid; v < 1024; v += 256)                      // 64x64
    stage16(&sW1[v * 4], &w1g[v * 4]);
  for (int v = tid; v < 4096; v += 256)                      // 64x256
    stage16(&sW2[v * 4], &w2g[v * 4]);

  const int nTiles = (E + 63) >> 6;
  for (int tile = blockIdx.x; tile < nTiles; tile += gridDim.x) {
    const int base = tile * 64;

    // Radial embedding + y1 + indices (one thread per edge).
    for (int e = tid; e < 64; e += 256) {
      const int ge = base + e;
      float len = 0.f, vx = 0.f, vy = 0.f, vz = 0.f;
      int s = 0, r = 0;
      if (ge < E) {
        len = elen[ge];
        vx = evec[ge * 3 + 0]; vy = evec[ge * 3 + 1]; vz = evec[ge * 3 + 2];
        s = send[ge]; r = recv[ge];
      }
      const float inv = rsqrtf(vx * vx + vy * vy + vz * vz + 1e-30f);
      const float s3 = 1.7320508f;
      sY[e * 3 + 0] = s3 * vx * inv;
      sY[e * 3 + 1] = s3 * vy * inv;
      sY[e * 3 + 2] = s3 * vz * inv;
      sS[e] = s; sR[e] = r;
      const float step = 5.0f / 9.0f;
#pragma unroll
      for (int i = 0; i < 8; ++i) {
        const float d = (len - step * (float)(i + 1)) / step;
        // sqrt(8)/1.12 * 1/sqrt(8) = 1/1.12
        sEmb[e * 8 + i] = expf(-d * d) * 0.89285713f;
      }
    }
    stage_wait();           // weights resident (no-op after first tile)
    __syncthreads();

    // --- GEMM 1: h1 = silu(emb' @ W0)*ACT/8  (64x8 @ 8x64) ---
    for (int t = wave * 2; t < wave * 2 + 2; ++t) {
      const int rt = t >> 2, ct = t & 3;
      v8f acc = {};
      gemm_tile<8, 1, 64, 8>(&sEmb[rt * 16 * 8], &sW0[ct * 16], acc, lane);
#pragma unroll
      for (int j = 0; j < 8; ++j)
        sH1[(rt * 16 + j + 8 * half) * 64 + ct * 16 + Mn] = silu_act(acc[j]) * 0.125f;
    }
    __syncthreads();

    // --- GEMM 2: h2 = silu(h1' @ W1)*ACT/8  (64x64 @ 64x64) ---
    for (int t = wave * 2; t < wave * 2 + 2; ++t) {
      const int rt = t >> 2, ct = t & 3;
      v8f acc = {};
      gemm_tile<64, 1, 64, 64>(&sH1[rt * 16 * 64], &sW1[ct * 16], acc, lane);
#pragma unroll
      for (int j = 0; j < 8; ++j)
        sH2[(rt * 16 + j + 8 * half) * 64 + ct * 16 + Mn] = silu_act(acc[j]) * 0.125f;
    }
    __syncthreads();

    // --- GEMM 3: gates = h2' @ W2  (64x64 @ 64x256) ---
    for (int t = wave * 8; t < wave * 8 + 8; ++t) {
      const int rt = t >> 4, ct = t & 15;
      v8f acc = {};
      gemm_tile<64, 1, 256, 64>(&sH2[rt * 16 * 64], &sW2[ct * 16], acc, lane);
#pragma unroll
      for (int j = 0; j < 8; ++j)
        sG[(rt * 16 + j + 8 * half) * 256 + ct * 16 + Mn] = acc[j];
    }
    __syncthreads();

    // --- Message build + scatter-add (coalesced over channel u) ---
    for (int it = 0; it < 16; ++it) {
      const int p = tid + it * 256;     // 64 edges * 64 channels
      const int e = p >> 6;
      const int u = p & 63;
      const int ge = base + e;
      if (ge < E) {
        const int s = sS[e];
        const int r = sR[e];
        const float g0 = sG[e * 256 + u];
        const float g1 = sG[e * 256 + 64 + u];
        const float g2 = sG[e * 256 + 128 + u];
        const float g3 = sG[e * 256 + 192 + u];
        const float xse = xs[(size_t)s * 64 + u];
        const float* xvp = &xv[(size_t)s * 192 + u * 3];
        const float x0 = xvp[0], x1 = xvp[1], x2 = xvp[2];
        const float y0 = sY[e * 3 + 0];
        const float y1 = sY[e * 3 + 1];
        const float y2 = sY[e * 3 + 2];
        const float m0 = g0 * xse;
        const float m1 = g1 * (x0 * y0 + x1 * y1 + x2 * y2) * 0.57735027f; // /sqrt(3)
        const float w2x = g2 * xse;
        float* mr = msg + (size_t)r * 512;
        atomicAdd(mr + u, m0);
        atomicAdd(mr + 64 + u, m1);
        atomicAdd(mr + 128 + u * 3 + 0, w2x * y0);
        atomicAdd(mr + 128 + u * 3 + 1, w2x * y1);
        atomicAdd(mr + 128 + u * 3 + 2, w2x * y2);
        atomicAdd(mr + 320 + u * 3 + 0, g3 * x0);
        atomicAdd(mr + 320 + u * 3 + 1, g3 * x1);
        atomicAdd(mr + 320 + u * 3 + 2, g3 * x2);
      }
    }
    __syncthreads();   // protect sEmb/sY/sS/sR/sG before next tile overwrites
  }
}

// ---------------------------------------------------------------------------
// Kernel 3: node "out" linears (weights raw; f = 1/(sqrt(128)*16) at D store).
//   out[n, c]        = sum_u msg[n, u]           * Wos[u, c] * f   (c<64)
//   out[n, 64+v*3+m] = sum_u msg[n, 128+u*3+m]   * Wov[u, v] * f
// ---------------------------------------------------------------------------
__global__ void k_node_out(const float* __restrict__ msg,
                           const float* __restrict__ wos,
                           const float* __restrict__ wov,
                           float* __restrict__ out, int N) {
  __shared__ float sM[16 * 512];
  __shared__ float sWs[128 * 64];
  __shared__ float sWv[128 * 64];
  const int tid = threadIdx.x;
  const int base = blockIdx.x * 16;

  for (int v = tid; v < 2048; v += 256) {          // 16 rows * 128 float4
    const int row = v >> 7;
    const int n = (base + row < N) ? (base + row) : (N - 1);
    stage16(&sM[v * 4], &msg[(size_t)n * 512 + (v & 127) * 4]);
  }
  for (int v = tid; v < 2048; v += 256) {          // 128x64 weights x2
    stage16(&sWs[v * 4], &wos[v * 4]);
    stage16(&sWv[v * 4], &wov[v * 4]);
  }
  stage_wait();
  __syncthreads();

  const int lane = tid & 31;
  const int wave = tid >> 5;
  const int Mn = lane & 15;
  const int half = lane >> 4;
  const float f = 0.00552427f;  // 1/(sqrt(128)*16)

  for (int t = wave * 2; t < wave * 2 + 2; ++t) {
    const int g = t >> 2;   // 0 = scalar out, 1..3 = vector component m = g-1
    const int c = t & 3;
    v8f acc = {};
    if (g == 0)
      gemm_tile<512, 1, 64, 128>(sM, &sWs[c * 16], acc, lane);
    else
      gemm_tile<512, 3, 64, 128>(&sM[128 + (g - 1)], &sWv[c * 16], acc, lane);
#pragma unroll
    for (int j = 0; j < 8; ++j) {
      const int n = base + j + 8 * half;
      if (n < N) {
        const int col = c * 16 + Mn;
        const float val = acc[j] * f;
        if (g == 0) out[(size_t)n * 256 + col] = val;
        else        out[(size_t)n * 256 + 64 + col * 3 + (g - 1)] = val;
      }
    }
  }
}

// ---------------------------------------------------------------------------
extern "C" void kernel_launch(void* const* d_in, const int* in_sizes, int n_in,
                              void* d_out, int out_size, void* d_ws, size_t ws_size,
                              hipStream_t stream) {
  const float* nf   = (const float*)d_in[0];
  const float* evec = (const float*)d_in[1];
  const float* elen = (const float*)d_in[2];
  const int*   snd  = (const int*)d_in[3];
  const int*   rcv  = (const int*)d_in[4];
  // d_in[5] = num_nodes (device scalar) — derived from in_sizes instead.
  const float* lus  = (const float*)d_in[6];
  const float* luv  = (const float*)d_in[7];
  const float* w0   = (const float*)d_in[8];
  const float* w1   = (const float*)d_in[9];
  const float* w2   = (const float*)d_in[10];
  const float* los  = (const float*)d_in[11];
  const float* lov  = (const float*)d_in[12];

  const int N = in_sizes[0] / 256;   // node_feats is (N, 256)
  const int E = in_sizes[2];         // edge_length is (E,)

  float* xs  = (float*)d_ws;                    // N*64
  float* xv  = xs + (size_t)N * 64;             // N*192
  float* msg = xv + (size_t)N * 192;            // N*512
  float* out = (float*)d_out;

  const int nTiles = (E + 63) / 64;
  const int edgeGrid = nTiles < 640 ? nTiles : 640;

  k_zero<<<1024, 256, 0, stream>>>(msg, N * 512);
  k_node_up<<<(N + 15) / 16, 256, 0, stream>>>(nf, lus, luv, xs, xv, N);
  k_edge<<<edgeGrid, 256, 0, stream>>>(evec, elen, snd, rcv, w0, w1, w2,
                                       xs, xv, msg, E);
  k_node_out<<<(N + 15) / 16, 256, 0, stream>>>(msg, los, lov, out, N);
}